// SAGE_59304908423843
// MI455X (gfx1250) — compile-verified
//
#include <hip/hip_runtime.h>
#include <hip/hip_bf16.h>

#define NNODES 100000
#define FDIM   128
#define NEDGE  1000000
#define NPRED  100000

typedef __attribute__((ext_vector_type(2))) float v2f;
typedef __attribute__((ext_vector_type(8))) float v8f;

// ---------------- zero workspace ----------------
__global__ void k_zero(float* __restrict__ p, long n) {
    long i = (long)blockIdx.x * blockDim.x + threadIdx.x;
    long stride = (long)gridDim.x * blockDim.x;
    for (; i < n; i += stride) p[i] = 0.0f;
}

// ---------------- scatter-add aggregation: one wave (32 lanes) per edge ----------------
__global__ void k_scatter(const float* __restrict__ feat,
                          const int* __restrict__ src,
                          const int* __restrict__ dst,
                          float* __restrict__ agg,
                          float* __restrict__ deg,
                          int nedges) {
    int gtid = blockIdx.x * blockDim.x + threadIdx.x;
    int wave = gtid >> 5;          // wave32
    int lane = gtid & 31;
    if (wave >= nedges) return;
    int s = src[wave];
    int d = dst[wave];
    const float4 v = ((const float4*)(feat + (long)s * FDIM))[lane];
    float* a = agg + (long)d * FDIM + lane * 4;
    atomicAdd(a + 0, v.x);
    atomicAdd(a + 1, v.y);
    atomicAdd(a + 2, v.z);
    atomicAdd(a + 3, v.w);
    if (lane == 0) atomicAdd(deg + d, 1.0f);
}

// ---------------- SAGE layer GEMM: hout = act(hin@Wself + (agg/deg)@Wneigh + b) ----------------
// Block = 256 threads = 8 waves. Block computes rows [16*bid, 16*bid+16) x all 128 cols.
// Each wave: one 16x16 C tile via V_WMMA_F32_16X16X4_F32, K swept 128 in steps of 4.
__global__ void k_sage_gemm(const float* __restrict__ hin,
                            const float* __restrict__ agg,
                            const float* __restrict__ deg,
                            const float* __restrict__ Wself,
                            const float* __restrict__ Wneigh,
                            const float* __restrict__ bias,
                            float* __restrict__ hout,
                            int do_relu) {
    __shared__ float sSelf[16 * FDIM];
    __shared__ float sMean[16 * FDIM];
    __shared__ float sInv[16];

    const int rowbase = blockIdx.x * 16;
    const int tid = threadIdx.x;

    if (tid < 16) {
        float dg = deg[rowbase + tid];
        sInv[tid] = 1.0f / fmaxf(dg, 1.0f);
    }
    __syncthreads();

    for (int i = tid; i < 16 * FDIM; i += 256) {
        int m = i >> 7;
        int k = i & 127;
        long row = rowbase + m;
        sSelf[i] = hin[row * FDIM + k];
        sMean[i] = agg[row * FDIM + k] * sInv[m];
    }
    __syncthreads();

    const int wv    = tid >> 5;
    const int lane  = tid & 31;
    const int ncol  = wv * 16 + (lane & 15);
    const int khalf = (lane >= 16) ? 2 : 0;
    const int mrow  = lane & 15;

    v8f c = {};
    for (int k0 = 0; k0 < FDIM; k0 += 4) {
        v2f a, b;
        // self contribution
        a.x = sSelf[mrow * FDIM + k0 + khalf];
        a.y = sSelf[mrow * FDIM + k0 + khalf + 1];
        b.x = Wself[(k0 + khalf) * FDIM + ncol];
        b.y = Wself[(k0 + khalf + 1) * FDIM + ncol];
        c = __builtin_amdgcn_wmma_f32_16x16x4_f32(false, a, false, b, (short)0, c, false, false);
        // mean-neighbor contribution
        a.x = sMean[mrow * FDIM + k0 + khalf];
        a.y = sMean[mrow * FDIM + k0 + khalf + 1];
        b.x = Wneigh[(k0 + khalf) * FDIM + ncol];
        b.y = Wneigh[(k0 + khalf + 1) * FDIM + ncol];
        c = __builtin_amdgcn_wmma_f32_16x16x4_f32(false, a, false, b, (short)0, c, false, false);
    }

    const float bv = bias[ncol];
    const int rowoff = (lane >= 16) ? 8 : 0;
    for (int r = 0; r < 8; ++r) {
        float val = c[r] + bv;
        if (do_relu) val = fmaxf(val, 0.0f);
        hout[(long)(rowbase + r + rowoff) * FDIM + ncol] = val;
    }
}

// ---------------- fused predictor: out[e] = relu((h[s]*h[d])@Wp1+bp1)@Wp2 + bp2 ----------------
__global__ void k_pred(const float* __restrict__ h,
                       const int* __restrict__ esrc,
                       const int* __restrict__ edst,
                       const float* __restrict__ Wp1,
                       const float* __restrict__ bp1,
                       const float* __restrict__ Wp2,
                       const float* __restrict__ bp2,
                       float* __restrict__ out) {
    __shared__ float sZ[16 * FDIM];
    __shared__ float sRed[8][16];
    __shared__ int sS[16], sD[16];

    const int eb = blockIdx.x * 16;
    const int tid = threadIdx.x;

    if (tid < 16) {
        sS[tid] = esrc[eb + tid];
        sD[tid] = edst[eb + tid];
    }
    __syncthreads();

    for (int i = tid; i < 16 * FDIM; i += 256) {
        int m = i >> 7;
        int k = i & 127;
        sZ[i] = h[(long)sS[m] * FDIM + k] * h[(long)sD[m] * FDIM + k];
    }
    __syncthreads();

    const int wv    = tid >> 5;
    const int lane  = tid & 31;
    const int ncol  = wv * 16 + (lane & 15);
    const int khalf = (lane >= 16) ? 2 : 0;
    const int mrow  = lane & 15;

    v8f c = {};
    for (int k0 = 0; k0 < FDIM; k0 += 4) {
        v2f a, b;
        a.x = sZ[mrow * FDIM + k0 + khalf];
        a.y = sZ[mrow * FDIM + k0 + khalf + 1];
        b.x = Wp1[(k0 + khalf) * FDIM + ncol];
        b.y = Wp1[(k0 + khalf + 1) * FDIM + ncol];
        c = __builtin_amdgcn_wmma_f32_16x16x4_f32(false, a, false, b, (short)0, c, false, false);
    }

    const float bv = bp1[ncol];
    const float w2 = Wp2[ncol];
    const int rowoff = (lane >= 16) ? 8 : 0;
    for (int r = 0; r < 8; ++r) {
        float val = fmaxf(c[r] + bv, 0.0f) * w2;   // relu then dot with Wp2
        // reduce over this wave's 16 columns (half-wave groups of 16 lanes)
        for (int off = 8; off >= 1; off >>= 1)
            val += __shfl_xor(val, off, 16);
        if ((lane & 15) == 0) sRed[wv][r + rowoff] = val;
    }
    __syncthreads();

    if (tid < 16) {
        float total = bp2[0];
        for (int w = 0; w < 8; ++w) total += sRed[w][tid];
        out[eb + tid] = total;
    }
}

extern "C" void kernel_launch(void* const* d_in, const int* in_sizes, int n_in,
                              void* d_out, int out_size, void* d_ws, size_t ws_size,
                              hipStream_t stream) {
    const float* x    = (const float*)d_in[0];
    const int*   es1  = (const int*)d_in[1];
    const int*   ed1  = (const int*)d_in[2];
    const int*   es2  = (const int*)d_in[3];
    const int*   ed2  = (const int*)d_in[4];
    const int*   ps   = (const int*)d_in[5];
    const int*   pd   = (const int*)d_in[6];
    const int*   ns   = (const int*)d_in[7];
    const int*   nd   = (const int*)d_in[8];
    const float* Ws1  = (const float*)d_in[9];
    const float* Wn1  = (const float*)d_in[10];
    const float* b1   = (const float*)d_in[11];
    const float* Ws2  = (const float*)d_in[12];
    const float* Wn2  = (const float*)d_in[13];
    const float* b2   = (const float*)d_in[14];
    const float* Wp1  = (const float*)d_in[15];
    const float* bp1  = (const float*)d_in[16];
    const float* Wp2  = (const float*)d_in[17];
    const float* bp2  = (const float*)d_in[18];
    float* out = (float*)d_out;

    float* agg = (float*)d_ws;
    float* deg = agg + (size_t)NNODES * FDIM;
    float* h1  = deg + NNODES;
    float* h2  = h1 + (size_t)NNODES * FDIM;

    const long aggdeg = (long)NNODES * FDIM + NNODES;
    const int scatter_blocks = (NEDGE * 32) / 256;   // one wave per edge
    const int gemm_blocks = NNODES / 16;             // 6250, exact
    const int pred_blocks = NPRED / 16;              // 6250, exact

    // Layer 1
    k_zero<<<2048, 256, 0, stream>>>(agg, aggdeg);
    k_scatter<<<scatter_blocks, 256, 0, stream>>>(x, es1, ed1, agg, deg, NEDGE);
    k_sage_gemm<<<gemm_blocks, 256, 0, stream>>>(x, agg, deg, Ws1, Wn1, b1, h1, 1);

    // Layer 2
    k_zero<<<2048, 256, 0, stream>>>(agg, aggdeg);
    k_scatter<<<scatter_blocks, 256, 0, stream>>>(h1, es2, ed2, agg, deg, NEDGE);
    k_sage_gemm<<<gemm_blocks, 256, 0, stream>>>(h1, agg, deg, Ws2, Wn2, b2, h2, 0);

    // Predictor (pos then neg), fused GEMM + reduce
    k_pred<<<pred_blocks, 256, 0, stream>>>(h2, ps, pd, Wp1, bp1, Wp2, bp2, out);
    k_pred<<<pred_blocks, 256, 0, stream>>>(h2, ns, nd, Wp1, bp1, Wp2, bp2, out + NPRED);
}